// HomogeneousPoissonEncoder_44186623541580
// MI455X (gfx1250) — compile-verified
//
#include <hip/hip_runtime.h>

#define STEPS   500
#define MTOT    65536u          // 32*2048 neurons
#define HHALF   16384000u       // STEPS*MTOT/2 = 250*MTOT (threefry iota split point)
#define BLOCK   256
#define TINYF   1.17549435e-38f

typedef unsigned v4u __attribute__((ext_vector_type(4)));
typedef int      v8i __attribute__((ext_vector_type(8)));
typedef int      v4i __attribute__((ext_vector_type(4)));

__device__ __forceinline__ unsigned rotl32(unsigned x, int r) {
    return (x << r) | (x >> (32 - r));
}

// Exact JAX threefry2x32 with key (0, 42); returns x0 if lo else x1.
__device__ __forceinline__ unsigned threefry_42(unsigned c0, unsigned c1, bool lo) {
    const unsigned k0 = 0u, k1 = 42u, k2 = 0x1BD11BF0u; // 0^42^0x1BD11BDA
    unsigned x0 = c0 + k0, x1 = c1 + k1;
#define TF_QR(r) { x0 += x1; x1 = rotl32(x1, r); x1 ^= x0; }
    TF_QR(13) TF_QR(15) TF_QR(26) TF_QR(6)   x0 += k1; x1 += k2 + 1u;
    TF_QR(17) TF_QR(29) TF_QR(16) TF_QR(24)  x0 += k2; x1 += k0 + 2u;
    TF_QR(13) TF_QR(15) TF_QR(26) TF_QR(6)   x0 += k0; x1 += k1 + 3u;
    TF_QR(17) TF_QR(29) TF_QR(16) TF_QR(24)  x0 += k1; x1 += k2 + 4u;
    TF_QR(13) TF_QR(15) TF_QR(26) TF_QR(6)   x0 += k2; x1 += k0 + 5u;
#undef TF_QR
    return lo ? x0 : x1;
}

// u[i, m] as produced by jax.random.uniform(key(42), (500,32,2048), minval=tiny)
__device__ __forceinline__ float uniform_at(unsigned i, unsigned m) {
    unsigned g  = i * MTOT + m;            // flat index into (S,B,N)
    bool     lo = g < HHALF;
    unsigned c0 = lo ? g : (g - HHALF);
    unsigned c1 = lo ? (g + HHALF) : g;
    unsigned bits = threefry_42(c0, c1, lo);
    float f = __uint_as_float((bits >> 9) | 0x3f800000u) - 1.0f;  // [0,1)
    return fmaxf(f, TINYF);
}

struct SharedBlk {
    float    tile[BLOCK];          // input intensities staged by TDM (LDS offset 0)
    unsigned bits[16 * BLOCK];     // 512-bit spike bitmask per thread, bank-conflict-free
};

__global__ void __launch_bounds__(BLOCK)
poisson_encode_kernel(const float* __restrict__ in, float* __restrict__ out) {
    __shared__ SharedBlk sh;
    const unsigned tid = threadIdx.x;
    const unsigned gm  = blockIdx.x * BLOCK + tid;

    // ---- Stage this block's 256 input floats into LDS via the Tensor Data Mover
    if (tid < 32) {  // one wave issues the DMA (EXEC all-ones in this wave)
        unsigned long long ga = (unsigned long long)(const void*)(in + blockIdx.x * BLOCK);
        // D# group 0: count=1 | lds_addr=0 | global_addr | type=2
        v4u g0 = { 1u,
                   0u,
                   (unsigned)(ga & 0xFFFFFFFFull),
                   (unsigned)((ga >> 32) & 0x1FFFFFFull) | (2u << 30) };
        // D# group 1: data_size=4B, tensor_dim0=256, tensor_dim1=1,
        //             tile_dim0=256, tile_dim1=1, tensor_dim0_stride=256
        v8i g1 = { (int)(2u << 16),     // data_size = 2 (4 bytes)
                   (int)(256u << 16),   // tensor_dim0[15:0] -> word1[31:16]
                   (int)(1u << 16),     // tensor_dim1[15:0] -> word2[31:16]
                   (int)(256u << 16),   // tile_dim0 -> word3[31:16]
                   1,                   // tile_dim1 -> word4[15:0]
                   256,                 // tensor_dim0_stride low32
                   0, 0 };
        v4i z4 = { 0, 0, 0, 0 };
#if defined(__clang_major__) && (__clang_major__ >= 23)
        v8i z8 = { 0, 0, 0, 0, 0, 0, 0, 0 };
        __builtin_amdgcn_tensor_load_to_lds(g0, g1, z4, z4, z8, 0);
#else
        __builtin_amdgcn_tensor_load_to_lds(g0, g1, z4, z4, 0);
#endif
        __builtin_amdgcn_s_wait_tensorcnt(0);
    }

    // Zero this thread's private bitmask words (layout bits[j*BLOCK+tid]: no bank conflicts)
#pragma unroll
    for (int j = 0; j < 16; ++j) sh.bits[j * BLOCK + tid] = 0u;

    __syncthreads();   // TDM tile visible to all waves

    // ---- Rate / lambda, matching reference op order exactly
    float x    = sh.tile[tid];
    float rate = 100.0f * x;
    rate       = rate / (1.0f - (rate * 1.0f) / 1000.0f);   // COMPENSATE, REFRAC=1ms
    float lam  = rate / 1000.0f;

    // ---- Spike-driven loop: consume exponential intervals in reference order
    if (lam > 0.0f) {
        float dnm  = fmaxf(lam, TINYF);
        unsigned i = 1;
        float u    = uniform_at(0u, gm);
        float next = (-logf(u)) / dnm + 1.0f;          // times[0]
        while (next <= 500.0f) {                        // bin = ceil(next)-1 in [0,499]
            int bin = (int)ceilf(next) - 1;
            atomicOr(&sh.bits[(unsigned)(bin >> 5) * BLOCK + tid], 1u << (bin & 31));
            if (i >= STEPS) break;                      // only 500 intervals exist
            float uu = uniform_at(i, gm); ++i;
            float e  = (-logf(uu)) / dnm;
            next     = next + (e + 1.0f);               // cumsum order = reference
        }
    }
    // Same-wave DS ops are in-order: safe to read own bitmask words back.

    // ---- Coalesced streaming store of 0.0/1.0 floats (131 MB total)
    for (int j = 0; j < 15; ++j) {                      // steps 0..479
        unsigned w = sh.bits[j * BLOCK + tid];
        float* po  = out + (size_t)(j * 32) * MTOT + gm;
#pragma unroll
        for (int b = 0; b < 32; ++b)
            po[(size_t)b * MTOT] = ((w >> b) & 1u) ? 1.0f : 0.0f;
    }
    {                                                   // steps 480..499 (20 bits)
        unsigned w = sh.bits[15 * BLOCK + tid];
        float* po  = out + (size_t)480 * MTOT + gm;
#pragma unroll
        for (int b = 0; b < 20; ++b)
            po[(size_t)b * MTOT] = ((w >> b) & 1u) ? 1.0f : 0.0f;
    }
}

extern "C" void kernel_launch(void* const* d_in, const int* in_sizes, int n_in,
                              void* d_out, int out_size, void* d_ws, size_t ws_size,
                              hipStream_t stream) {
    const float* in = (const float*)d_in[0];
    float* out      = (float*)d_out;
    dim3 grid(MTOT / BLOCK), block(BLOCK);
    hipLaunchKernelGGL(poisson_encode_kernel, grid, block, 0, stream, in, out);
}